// ContextualLSTM_23235773071402
// MI455X (gfx1250) — compile-verified
//
#include <hip/hip_runtime.h>
#include <math.h>

// CDNA5 / gfx1250, wave32. fp32 WMMA (V_WMMA_F32_16X16X4_F32) throughout to
// match the fp32 reference numerics. Data movement uses the CDNA5 async paths:
// GLOBAL_LOAD_ASYNC_TO_LDS_B128 (ASYNCcnt) for GEMM tile staging and the
// Tensor Data Mover (TENSORcnt) for the recurrent-weight LDS preload.

typedef float v2f __attribute__((ext_vector_type(2)));
typedef float v8f __attribute__((ext_vector_type(8)));
typedef unsigned int v4u __attribute__((ext_vector_type(4)));
typedef int v4i __attribute__((ext_vector_type(4)));
typedef int v8i __attribute__((ext_vector_type(8)));

#define V_   50000
#define E_   512
#define H_   256
#define B_   32
#define S_   1024
#define G4H  1024   // 4*H

__device__ __forceinline__ float sigf(float x) { return 1.0f / (1.0f + __expf(-x)); }

// Async DMA: global -> LDS, 16B per lane, tracked by ASYNCcnt.
__device__ __forceinline__ void async_ld_b128(void* lds_ptr, const void* gptr) {
    asm volatile("global_load_async_to_lds_b128 %0, %1, off"
                 :: "v"((unsigned)(size_t)lds_ptr), "v"(gptr)
                 : "memory");
}
__device__ __forceinline__ void wait_async0() {
    asm volatile("s_wait_asynccnt 0x0" ::: "memory");
}

// TDM 2-D tile load: tile_rows x row_dwords (4B elements), row-major source
// with stride row_dwords, LDS destination padded +2 DWORDs per 256 DWORDs
// (pad_interval enum 7, pad_amount enum 1) -> 258-float LDS rows.
__device__ __forceinline__ void tdm_load_2d(unsigned lds_addr, const void* gaddr,
                                            unsigned tile_rows, unsigned row_dwords) {
    unsigned long long ga = (unsigned long long)gaddr;
    v4u g0;
    g0.x = 0x1u;                                   // count=1, user mode
    g0.y = lds_addr;                               // D# lds_addr (bytes)
    g0.z = (unsigned)ga;                           // global_addr[31:0]
    g0.w = (unsigned)(ga >> 32) | (2u << 30);      // global_addr[56:32] | type=2
    v8i g1;
    g1[0] = (int)((2u << 16) | (1u << 20) | (7u << 22) | (1u << 25));
    //        data_size=4B  | pad_enable | pad_int=256DW | pad_amt=2DW ; wg_mask=0
    g1[1] = (int)(row_dwords << 16);               // tensor_dim0[15:0]
    g1[2] = (int)((tile_rows & 0xFFFFu) << 16);    // dim0 hi=0 | tensor_dim1 lo
    g1[3] = (int)(row_dwords << 16);               // dim1 hi=0 | tile_dim0
    g1[4] = (int)(tile_rows & 0xFFFFu);            // tile_dim1 | tile_dim2=0
    g1[5] = (int)row_dwords;                       // tensor_dim0_stride lo32
    g1[6] = 0;                                     // stride hi | dim1_stride lo
    g1[7] = 0;
    v4i z4 = {0, 0, 0, 0};                         // <=2D: groups 2/3 unused
    v8i z8 = {0, 0, 0, 0, 0, 0, 0, 0};
    // 6-arg form on this toolchain (therock-10.0 headers / clang-23 lane).
    __builtin_amdgcn_tensor_load_to_lds(g0, g1, z4, z4, z8, 0);
}

// ---------------------------------------------------------------------------
// Phase 0: rank[b][t] = exclusive prefix count of fmask, counts[b] = total.
// ---------------------------------------------------------------------------
__global__ void rank_kernel(const int* __restrict__ fmask,
                            int* __restrict__ rank, int* __restrict__ counts) {
    int b = threadIdx.x;
    if (b >= B_) return;
    int r = 0;
    for (int t = 0; t < S_; ++t) {
        rank[b * S_ + t] = r;
        r += (fmask[b * S_ + t] != 0);
    }
    counts[b] = r;
}

// ---------------------------------------------------------------------------
// Phase 1: xg[m][c] = sum_k emb[tok(m)][k] * Wih[c][k] + bih[c] + bhh[c]
// WG tile 64x64, 8 waves, 2 WMMA subtiles/wave. K staged 16 at a time through
// double-buffered LDS tiles filled by GLOBAL_LOAD_ASYNC_TO_LDS_B128 so DMA of
// chunk c+1 overlaps WMMA on chunk c.
// ---------------------------------------------------------------------------
__global__ __launch_bounds__(256)
void embed_gemm_kernel(const int* __restrict__ inputs, const int* __restrict__ seq_len,
                       const float* __restrict__ emb, const float* __restrict__ Wih,
                       const float* __restrict__ bih, const float* __restrict__ bhh,
                       float* __restrict__ xg, int reverse) {
    __shared__ __align__(16) float As[2][64][20];  // [buf][row m][k]  (+pad)
    __shared__ __align__(16) float Bs[2][64][20];  // [buf][col n][k]  (+pad)
    __shared__ int tok[64];

    const int m0   = blockIdx.x * 64;   // 512 blocks over M=32768
    const int n0   = blockIdx.y * 64;   // 16 blocks over N=1024
    const int tid  = threadIdx.x;
    const int lane = tid & 31;
    const int wave = tid >> 5;
    const int lo   = lane & 15;
    const int hi   = lane >> 4;         // K-half select per ISA frag layout

    if (tid < 64) {
        int m = m0 + tid;
        int b = m >> 10;                // / S_
        int s = m & (S_ - 1);
        int tk;
        if (reverse) {
            int L = seq_len[b];
            tk = (s < L) ? inputs[b * S_ + (L - 1 - s)] : 0;
        } else {
            tk = inputs[b * S_ + s];
        }
        tok[tid] = tk;
    }
    __syncthreads();

    const int mt  = wave & 3;           // M subtile 0..3
    const int ntb = (wave >> 2) << 1;   // N subtile base: 0 or 2
    const int sr  = tid >> 2;           // staging row 0..63
    const int sc4 = (tid & 3) << 2;     // staging k offset 0,4,8,12
    v8f acc0 = {}; v8f acc1 = {};

    // Prologue: DMA chunk 0 into buffer 0.
    async_ld_b128(&As[0][sr][sc4], emb + (size_t)tok[sr] * E_ + sc4);
    async_ld_b128(&Bs[0][sr][sc4], Wih + (size_t)(n0 + sr) * E_ + sc4);
    wait_async0();
    __syncthreads();

    for (int c = 0; c < E_ / 16; ++c) {
        if (c + 1 < E_ / 16) {  // DMA next chunk while computing this one
            int k0 = (c + 1) * 16;
            int nb = (c + 1) & 1;
            async_ld_b128(&As[nb][sr][sc4], emb + (size_t)tok[sr] * E_ + k0 + sc4);
            async_ld_b128(&Bs[nb][sr][sc4], Wih + (size_t)(n0 + sr) * E_ + k0 + sc4);
        }
        const int bs = c & 1;
        #pragma unroll
        for (int kk = 0; kk < 16; kk += 4) {
            // A 16x4 frag: lanes 0-15 rows, VGPR0/1 = K{0,1}; lanes 16-31 K{2,3}
            v2f a, b0, b1;
            a.x  = As[bs][mt * 16 + lo][kk + hi * 2 + 0];
            a.y  = As[bs][mt * 16 + lo][kk + hi * 2 + 1];
            // B 4x16 frag from [n][k] tile: B[k][n] = Wih[n0+n][k0+k]
            b0.x = Bs[bs][ntb * 16 + lo][kk + hi * 2 + 0];
            b0.y = Bs[bs][ntb * 16 + lo][kk + hi * 2 + 1];
            b1.x = Bs[bs][(ntb + 1) * 16 + lo][kk + hi * 2 + 0];
            b1.y = Bs[bs][(ntb + 1) * 16 + lo][kk + hi * 2 + 1];
            acc0 = __builtin_amdgcn_wmma_f32_16x16x4_f32(false, a, false, b0,
                                                         (short)0, acc0, false, false);
            acc1 = __builtin_amdgcn_wmma_f32_16x16x4_f32(false, a, false, b1,
                                                         (short)0, acc1, false, false);
        }
        wait_async0();      // next buffer's DMA has landed
        __syncthreads();    // everyone done reading current buffer
    }

    #pragma unroll
    for (int v = 0; v < 8; ++v) {
        int row = m0 + mt * 16 + v + hi * 8;
        int c0  = n0 + ntb * 16 + lo;
        int c1  = n0 + (ntb + 1) * 16 + lo;
        xg[(size_t)row * G4H + c0] = acc0[v] + bih[c0] + bhh[c0];
        xg[(size_t)row * G4H + c1] = acc1[v] + bih[c1] + bhh[c1];
    }
}

// ---------------------------------------------------------------------------
// Phase 2: sequential LSTM scan. 16 WGs; WG g owns gate columns
// {q*256 + g*16 .. +16 | q=0..3} == h columns [g*16, g*16+16).
// W_hh slice (64x256) preloaded to LDS by 4 TDM descriptors; h exchanged via
// L2 (double-buffered) under a device-scope arrive/spin barrier. Two WMMA
// accumulator chains halve the per-step XDL critical path.
// ---------------------------------------------------------------------------
__global__ __launch_bounds__(256)
void lstm_scan_kernel(const float* __restrict__ xg, const float* __restrict__ Whh,
                      const int* __restrict__ fmask, const int* __restrict__ rank,
                      const int* __restrict__ counts,
                      float* __restrict__ hbuf,          // [2][B][H]
                      unsigned int* __restrict__ ctr,
                      float* __restrict__ out, int osl, int dir) {
    __shared__ __align__(16) float Ws[64][258];   // stride%64==2: conflict-free
    __shared__ __align__(16) float hls[32][258];
    __shared__ float gates[32][68];
    __shared__ float cst[32][16];

    const int g    = blockIdx.x;     // 0..15 slice
    const int tid  = threadIdx.x;
    const int lane = tid & 31;
    const int wave = tid >> 5;
    const int lo   = lane & 15;
    const int hi   = lane >> 4;
    const int mt   = wave & 1;       // 2 M tiles (batch 0..31)
    const int nt   = wave >> 1;      // 4 N tiles == gate q

    // TDM preload: Ws[q*16 + r][k] = Whh[q*H + g*16 + r][k], one descriptor
    // per gate block; LDS pad descriptor recreates the 258-float row stride.
    if (wave == 0) {
        #pragma unroll
        for (int q = 0; q < 4; ++q) {
            unsigned lws = (unsigned)(size_t)(&Ws[q * 16][0]);
            tdm_load_2d(lws, Whh + (size_t)(q * H_ + g * 16) * H_, 16, H_);
        }
        __builtin_amdgcn_s_wait_tensorcnt(0);
    }
    for (int i = tid; i < 32 * 16; i += 256) cst[i >> 4][i & 15] = 0.0f;
    for (int i = tid; i < 32 * 256; i += 256) hls[i >> 8][i & 255] = 0.0f;
    __syncthreads();

    const int nl = nt * 16 + lo;                 // local gate column 0..63
    const int cg = nt * H_ + g * 16 + lo;        // global gate column

    for (int t = 0; t < S_; ++t) {
        v8f acc0, acc1 = {};
        #pragma unroll
        for (int v = 0; v < 8; ++v) {            // C preload = xg fragment
            int m = mt * 16 + v + hi * 8;
            acc0[v] = xg[((size_t)m * S_ + t) * G4H + cg];
        }
        if (t + 1 < S_) {   // hide next step's xg latency (global_prefetch_b8)
            #pragma unroll
            for (int v = 0; v < 8; ++v) {
                int m = mt * 16 + v + hi * 8;
                __builtin_prefetch(&xg[((size_t)m * S_ + (t + 1)) * G4H + cg], 0, 1);
            }
        }
        // gates += h(32x256) @ Whh_slice^T : 64 WMMAs, two independent chains
        #pragma unroll 4
        for (int k = 0; k < H_; k += 8) {
            v2f a0, b0, a1, b1;
            a0.x = hls[mt * 16 + lo][k + hi * 2 + 0];
            a0.y = hls[mt * 16 + lo][k + hi * 2 + 1];
            b0.x = Ws[nl][k + hi * 2 + 0];
            b0.y = Ws[nl][k + hi * 2 + 1];
            a1.x = hls[mt * 16 + lo][k + 4 + hi * 2 + 0];
            a1.y = hls[mt * 16 + lo][k + 4 + hi * 2 + 1];
            b1.x = Ws[nl][k + 4 + hi * 2 + 0];
            b1.y = Ws[nl][k + 4 + hi * 2 + 1];
            acc0 = __builtin_amdgcn_wmma_f32_16x16x4_f32(false, a0, false, b0,
                                                         (short)0, acc0, false, false);
            acc1 = __builtin_amdgcn_wmma_f32_16x16x4_f32(false, a1, false, b1,
                                                         (short)0, acc1, false, false);
        }
        #pragma unroll
        for (int v = 0; v < 8; ++v)
            gates[mt * 16 + v + hi * 8][nl] = acc0[v] + acc1[v];
        __syncthreads();

        // LSTM cell on this WG's 32x16 (batch x hcol) slice; 2 pairs/thread.
        #pragma unroll
        for (int r = 0; r < 2; ++r) {
            int p = tid + r * 256;
            int m = p >> 4, j = p & 15;
            float ig = gates[m][j];
            float fg = gates[m][16 + j];
            float gg = gates[m][32 + j];
            float og = gates[m][48 + j];
            float c  = sigf(fg) * cst[m][j] + sigf(ig) * tanhf(gg);
            cst[m][j] = c;
            float h  = sigf(og) * tanhf(c);
            hbuf[(((size_t)(t & 1)) * B_ + m) * H_ + g * 16 + j] = h;
            if (fmask[m * S_ + t]) {
                int rk = rank[m * S_ + t];
                int jo = dir ? (counts[m] - 1 - rk) : rk;
                int co = dir ? (H_ + g * 16 + j) : (g * 16 + j);
                if (jo >= 0 && jo < osl)
                    out[((size_t)m * osl + jo) * (2 * H_) + co] = h;
            }
        }
        // Device-scope arrive/spin barrier across the 16 WGs.
        __threadfence();
        __syncthreads();
        if (tid == 0) {
            atomicAdd(ctr, 1u);
            unsigned int target = 16u * (unsigned)(t + 1);
            while (__hip_atomic_load(ctr, __ATOMIC_ACQUIRE,
                                     __HIP_MEMORY_SCOPE_AGENT) < target) {
                __builtin_amdgcn_s_sleep(1);
            }
        }
        __syncthreads();
        __builtin_amdgcn_fence(__ATOMIC_ACQUIRE, "agent");

        if (t + 1 < S_) { // gather full h_t for next step (slot t&1)
            for (int i = tid; i < (B_ * H_) / 4; i += 256) {
                int m  = (i << 2) >> 8;
                int k4 = (i << 2) & 255;
                float4 v = *(const float4*)(hbuf + (((size_t)(t & 1)) * B_ + m) * H_ + k4);
                hls[m][k4 + 0] = v.x; hls[m][k4 + 1] = v.y;
                hls[m][k4 + 2] = v.z; hls[m][k4 + 3] = v.w;
            }
            __syncthreads();
        }
    }
}

// ---------------------------------------------------------------------------
extern "C" void kernel_launch(void* const* d_in, const int* in_sizes, int n_in,
                              void* d_out, int out_size, void* d_ws, size_t ws_size,
                              hipStream_t stream) {
    const int*   inputs  = (const int*)d_in[0];
    const int*   seqlen  = (const int*)d_in[1];
    const int*   fmask   = (const int*)d_in[2];
    // d_in[3] bmask: unused by the reference computation
    const float* emb     = (const float*)d_in[5];
    const float* f_Wih   = (const float*)d_in[6];
    const float* f_Whh   = (const float*)d_in[7];
    const float* f_bih   = (const float*)d_in[8];
    const float* f_bhh   = (const float*)d_in[9];
    const float* b_Wih   = (const float*)d_in[10];
    const float* b_Whh   = (const float*)d_in[11];
    const float* b_bih   = (const float*)d_in[12];
    const float* b_bhh   = (const float*)d_in[13];
    float* out = (float*)d_out;

    const int osl = out_size / (B_ * 2 * H_);   // out_seq_length

    // Workspace carve-up (xg reused across directions): ~128.2 MB needed.
    char*  w   = (char*)d_ws;
    size_t off = 0;
    float* xg    = (float*)(w + off); off += (size_t)B_ * S_ * G4H * 4;  // 128 MB
    float* hbuf  = (float*)(w + off); off += (size_t)2 * B_ * H_ * 4;    // 64 KB
    int*   rank  = (int*)(w + off);   off += (size_t)B_ * S_ * 4;        // 128 KB
    int*   cnts  = (int*)(w + off);   off += 256;
    unsigned int* ctr = (unsigned int*)(w + off); off += 256;
    if (ws_size < off) return;  // workspace too small; nothing sane to do

    (void)hipMemsetAsync(d_out, 0, (size_t)out_size * sizeof(float), stream);
    rank_kernel<<<1, 32, 0, stream>>>(fmask, rank, cnts);

    // Forward direction
    embed_gemm_kernel<<<dim3(512, 16), 256, 0, stream>>>(
        inputs, seqlen, emb, f_Wih, f_bih, f_bhh, xg, 0);
    (void)hipMemsetAsync(ctr, 0, 256, stream);
    lstm_scan_kernel<<<16, 256, 0, stream>>>(
        xg, f_Whh, fmask, rank, cnts, hbuf, ctr, out, osl, 0);

    // Backward direction (time-reversed gather), reuses xg/hbuf
    embed_gemm_kernel<<<dim3(512, 16), 256, 0, stream>>>(
        inputs, seqlen, emb, b_Wih, b_bih, b_bhh, xg, 1);
    (void)hipMemsetAsync(ctr, 0, 256, stream);
    lstm_scan_kernel<<<16, 256, 0, stream>>>(
        xg, b_Whh, fmask, rank, cnts, hbuf, ctr, out, osl, 1);
}